// yolo_detect_target_22926535426318
// MI455X (gfx1250) — compile-verified
//
#include <hip/hip_runtime.h>

#define CONF 0.25f

typedef float v2f __attribute__((ext_vector_type(2)));
typedef float v8f __attribute__((ext_vector_type(8)));

// ---- wave (32-lane) max reduction via lane permutes ----
__device__ __forceinline__ float wave_max32(float v) {
#pragma unroll
    for (int m = 16; m >= 1; m >>= 1)
        v = fmaxf(v, __shfl_xor(v, m, 32));
    return v;
}

// ---- wave (32-lane) sum reduction via V_WMMA_F32_16X16X4_F32 ----
// A-matrix 16x4 layout (ISA 7.12.2): lane m (0-15) supplies A[m][0..1],
// lane m+16 supplies A[m][2..3]. With a = {p, 0}:
//   A[m][0] = p_m, A[m][2] = p_{m+16}, others 0.
// B = all ones  =>  D[m][n] = p_m + p_{m+16} for every n.
// C/D layout: lane l<16 holds D[r][l] in d[r] (r=0..7), lane l>=16 holds
// D[r+8][l-16]. Summing d[0..7] in-lane plus the xor-16 partner gives the
// full 32-lane sum in every lane. Requires EXEC == all ones.
__device__ __forceinline__ float wave_sum32_wmma(float p) {
    v2f a; a[0] = p;    a[1] = 0.0f;
    v2f b; b[0] = 1.0f; b[1] = 1.0f;
    v8f c = {};
    v8f d = __builtin_amdgcn_wmma_f32_16x16x4_f32(
        /*neg_a=*/false, a, /*neg_b=*/false, b,
        /*c_mod=*/(short)0, c, /*reuse_a=*/false, /*reuse_b=*/false);
    float t = 0.0f;
#pragma unroll
    for (int i = 0; i < 8; ++i) t += d[i];
    t += __shfl_xor(t, 16, 32);
    return t;
}

// ---- kernel 0: init workspace scalars ----
__global__ void k_init(unsigned int* ws_u, unsigned int num) {
    if (threadIdx.x == 0 && blockIdx.x == 0) {
        ws_u[0] = num;  // K = first break index, init to num
        ws_u[1] = 0u;   // transformed max bits
    }
}

// ---- kernel 1: find first row in [0,num) whose rowmax < CONF ----
__global__ __launch_bounds__(256) void k_find_break(
    const float* __restrict__ post, int C, int num, unsigned int* __restrict__ Kp) {
    const int lane = threadIdx.x & 31;
    const int row  = blockIdx.x * (blockDim.x >> 5) + (threadIdx.x >> 5);
    if (row >= num) return;
    const float* r = post + (size_t)row * (size_t)C;
    float m = -INFINITY;
    for (int j = lane; j < C; j += 32) m = fmaxf(m, r[j]);
    m = wave_max32(m);
    if (lane == 0 && !(m >= CONF)) atomicMin(Kp, (unsigned int)row);
}

// ---- kernel 2: deterministic partial sums of (rowmax + boxsum) for i < K ----
// Score part: one wave per row (coalesced b32 lane loads, branch-free select).
// Box part:   one thread per row, float4 loads (fully coalesced, branch-free).
// Both loops reconverge before the WMMA wave reduction (EXEC all-ones there).
__global__ __launch_bounds__(256) void k_prefix_sum(
    const float* __restrict__ post, const float* __restrict__ boxes, int C,
    const unsigned int* __restrict__ Kp, float* __restrict__ partials) {
    __shared__ float lds[8];
    const unsigned int K = *Kp;                 // uniform across grid
    const int lane   = threadIdx.x & 31;
    const int waveIB = threadIdx.x >> 5;
    const unsigned int gwave    = blockIdx.x * (blockDim.x >> 5) + waveIB;
    const unsigned int nwaves   = gridDim.x * (blockDim.x >> 5);
    const unsigned int gtid     = blockIdx.x * blockDim.x + threadIdx.x;
    const unsigned int nthreads = gridDim.x * blockDim.x;

    float acc = 0.0f;

    // scores: wave-uniform loop, no divergent branches inside
    for (unsigned int i = gwave; i < K; i += nwaves) {
        const float* r = post + (size_t)i * (size_t)C;
        float m = -INFINITY;
        for (int j = lane; j < C; j += 32) m = fmaxf(m, r[j]);
        m = wave_max32(m);                      // all lanes hold rowmax
        acc += (lane == 0) ? m : 0.0f;          // cndmask, no exec branch
    }

    // boxes: sum of first K rows == flat sum; one float4 row per thread
    const float4* b4 = (const float4*)boxes;
    for (unsigned int i = gtid; i < K; i += nthreads) {
        float4 v = b4[i];
        acc += (v.x + v.y) + (v.z + v.w);
    }

    // full-EXEC here: blockDim = 256, no early returns
    float wtot = wave_sum32_wmma(acc);
    if (lane == 0) lds[waveIB] = wtot;
    __syncthreads();
    if (threadIdx.x == 0) {
        float s = 0.0f;
        for (int w = 0; w < 8; ++w) s += lds[w];         // fixed order
        partials[blockIdx.x] = s;
    }
}

// ---- kernel 3: fallback global max over all N*C elements (only if K==0) ----
__device__ __forceinline__ unsigned int float_to_ordered(float f) {
    unsigned int b = __float_as_uint(f);
    return (b & 0x80000000u) ? ~b : (b | 0x80000000u);
}

__global__ __launch_bounds__(256) void k_global_max(
    const float* __restrict__ post, long long n_total,
    const unsigned int* __restrict__ Kp, unsigned int* __restrict__ maxbits) {
    if (*Kp != 0u) return;                               // common case: skip
    __shared__ float lds[8];
    const long long n4 = n_total >> 2;
    const float4* p = (const float4*)post;
    long long idx    = (long long)blockIdx.x * blockDim.x + threadIdx.x;
    long long stride = (long long)gridDim.x * blockDim.x;
    float m = -INFINITY;
    for (long long i = idx; i < n4; i += stride) {
        __builtin_prefetch(&p[i + stride], 0, 0);        // global_prefetch_b8
        float4 v = p[i];
        m = fmaxf(m, fmaxf(fmaxf(v.x, v.y), fmaxf(v.z, v.w)));
    }
    if (idx == 0)                                        // scalar tail
        for (long long i = n4 << 2; i < n_total; ++i) m = fmaxf(m, post[i]);
    m = wave_max32(m);
    const int lane = threadIdx.x & 31, waveIB = threadIdx.x >> 5;
    if (lane == 0) lds[waveIB] = m;
    __syncthreads();
    if (threadIdx.x == 0) {
        float s = lds[0];
        for (int w = 1; w < 8; ++w) s = fmaxf(s, lds[w]);
        atomicMax(maxbits, float_to_ordered(s));         // order-independent
    }
}

// ---- kernel 4: finalize scalar output ----
__global__ void k_finalize(const unsigned int* __restrict__ ws_u,
                           const float* __restrict__ partials, int nparts,
                           float* __restrict__ out) {
    if (threadIdx.x == 0 && blockIdx.x == 0) {
        const unsigned int K = ws_u[0];
        if (K > 0u) {
            float s = 0.0f;
            for (int i = 0; i < nparts; ++i) s += partials[i];  // fixed order
            out[0] = s;
        } else {
            unsigned int b = ws_u[1];
            unsigned int bits = (b & 0x80000000u) ? (b ^ 0x80000000u) : ~b;
            out[0] = __uint_as_float(bits);
        }
    }
}

extern "C" void kernel_launch(void* const* d_in, const int* in_sizes, int n_in,
                              void* d_out, int out_size, void* d_ws, size_t ws_size,
                              hipStream_t stream) {
    const float* post  = (const float*)d_in[0];   // [N, C] f32
    const float* boxes = (const float*)d_in[1];   // [N, 4] f32
    const long long NC = (long long)in_sizes[0];
    const int N = in_sizes[1] / 4;
    const int C = (int)(NC / (long long)N);       // 80
    int num = (int)((double)N * 0.02);            // matches int(N*RATIO)
    if (num < 1) num = 1;
    if (num > N) num = N;

    unsigned int* ws_u = (unsigned int*)d_ws;     // [0]=K, [1]=maxbits
    float* partials    = (float*)d_ws + 2;        // NB floats
    float* out         = (float*)d_out;

    const int NB = 80;                            // partial-sum blocks

    k_init<<<1, 32, 0, stream>>>(ws_u, (unsigned int)num);

    const int wavesPerBlock = 8;                  // 256 threads
    const int blocks1 = (num + wavesPerBlock - 1) / wavesPerBlock;
    k_find_break<<<blocks1, 256, 0, stream>>>(post, C, num, ws_u);

    k_prefix_sum<<<NB, 256, 0, stream>>>(post, boxes, C, ws_u, partials);

    k_global_max<<<2048, 256, 0, stream>>>(post, NC, ws_u, ws_u + 1);

    k_finalize<<<1, 32, 0, stream>>>(ws_u, partials, NB, out);
}